// Granule_472446403038
// MI455X (gfx1250) — compile-verified
//
#include <hip/hip_runtime.h>

typedef float v2f __attribute__((ext_vector_type(2)));
typedef float v8f __attribute__((ext_vector_type(8)));

#define DD   512
#define NPRM 10
#define BB   512
#define FSCALE 0.70710678118654752440f

// D = A(16x4) * B(4x16) + C, f32 WMMA. A-frag: lanes 0-15 hold K=0(v0),1(v1);
// lanes 16-31 hold K=2(v0),3(v1). B-frag: v0 = B[k0][n], v1 = B[k0+1][n], n=lane&15,
// k0 = 2*(lane>>4). C/D: vgpr i -> row i (lanes 0-15) / row i+8 (lanes 16-31), col = lane&15.
__device__ __forceinline__ v8f wmma4(v2f a, v2f b, v8f c) {
  return __builtin_amdgcn_wmma_f32_16x16x4_f32(false, a, false, b, (short)0, c, false, false);
}

// ---------------- C[M,512] = alpha * (A[M,K] @ W[512,K]^T) + bias ----------------
// W row stride = ldw (supports net_w1 column-half views). grid.x = ceil(M/16), block 256.
__global__ void gemm_at512(const float* __restrict__ A, int M, int lda,
                           const float* __restrict__ W, int ldw, int Kdim,
                           const float* __restrict__ bias, float alpha,
                           float* __restrict__ C) {
  const int tid  = threadIdx.x;
  const int wave = tid >> 5, lane = tid & 31;
  const int hi   = lane >> 4, lr = lane & 15;
  const int rBase = blockIdx.x * 16;
  const int nBase = wave * 64;

  int rowA = rBase + lr;                 // clamp so loads stay valid; stores are guarded
  if (rowA >= M) rowA = M - 1;

  // incrementing pointers -> immediate-offset addressing in the loop
  const float* __restrict__ ap  = A + (size_t)rowA * lda + 2 * hi;
  const float* __restrict__ wp0 = W + (size_t)(nBase + lr) * ldw + 2 * hi;
  const float* __restrict__ wp1 = wp0 + (size_t)16 * ldw;
  const float* __restrict__ wp2 = wp0 + (size_t)32 * ldw;
  const float* __restrict__ wp3 = wp0 + (size_t)48 * ldw;

  v8f acc0 = {0,0,0,0,0,0,0,0}, acc1 = acc0, acc2 = acc0, acc3 = acc0;

#pragma unroll 2
  for (int kk = 0; kk < Kdim; kk += 4) {
    v2f a;  a.x  = ap[0];  a.y  = ap[1];  ap  += 4;
    v2f b0; b0.x = wp0[0]; b0.y = wp0[1]; wp0 += 4;
    v2f b1; b1.x = wp1[0]; b1.y = wp1[1]; wp1 += 4;
    v2f b2; b2.x = wp2[0]; b2.y = wp2[1]; wp2 += 4;
    v2f b3; b3.x = wp3[0]; b3.y = wp3[1]; wp3 += 4;

    acc0 = wmma4(a, b0, acc0);
    acc1 = wmma4(a, b1, acc1);
    acc2 = wmma4(a, b2, acc2);
    acc3 = wmma4(a, b3, acc3);
  }

#pragma unroll
  for (int i = 0; i < 8; ++i) {
    const int row = rBase + i + 8 * hi;
    if (row < M) {
      int col = nBase + lr;
      float bv;
      bv = bias ? bias[col] : 0.f; C[(size_t)row * DD + col] = alpha * acc0[i] + bv; col += 16;
      bv = bias ? bias[col] : 0.f; C[(size_t)row * DD + col] = alpha * acc1[i] + bv; col += 16;
      bv = bias ? bias[col] : 0.f; C[(size_t)row * DD + col] = alpha * acc2[i] + bv; col += 16;
      bv = bias ? bias[col] : 0.f; C[(size_t)row * DD + col] = alpha * acc3[i] + bv;
    }
  }
}

// ---------------- C[M,512] = A[M,K] @ Bm[K,512] (row-major B), used for attn @ v --------
__global__ void gemm_ab512(const float* __restrict__ A, int M, int lda,
                           const float* __restrict__ Bm, int ldb, int Kdim,
                           float* __restrict__ C) {
  const int tid  = threadIdx.x;
  const int wave = tid >> 5, lane = tid & 31;
  const int hi   = lane >> 4, lr = lane & 15;
  const int rBase = blockIdx.x * 16;
  const int nBase = wave * 64;

  int rowA = rBase + lr;
  if (rowA >= M) rowA = M - 1;
  const float* __restrict__ ap = A + (size_t)rowA * lda + 2 * hi;
  const float* __restrict__ bp = Bm + (size_t)2 * hi * ldb + nBase + lr;

  v8f acc0 = {0,0,0,0,0,0,0,0}, acc1 = acc0, acc2 = acc0, acc3 = acc0;

  for (int kk = 0; kk < Kdim; kk += 4) {
    v2f a; a.x = ap[0]; a.y = ap[1]; ap += 4;

    v2f b0; b0.x = bp[0];  b0.y = bp[ldb];
    v2f b1; b1.x = bp[16]; b1.y = bp[16 + ldb];
    v2f b2; b2.x = bp[32]; b2.y = bp[32 + ldb];
    v2f b3; b3.x = bp[48]; b3.y = bp[48 + ldb];
    bp += (size_t)4 * ldb;

    acc0 = wmma4(a, b0, acc0);
    acc1 = wmma4(a, b1, acc1);
    acc2 = wmma4(a, b2, acc2);
    acc3 = wmma4(a, b3, acc3);
  }

#pragma unroll
  for (int i = 0; i < 8; ++i) {
    const int row = rBase + i + 8 * hi;
    if (row < M) {
      int col = nBase + lr;
      C[(size_t)row * DD + col]      = acc0[i];
      C[(size_t)row * DD + col + 16] = acc1[i];
      C[(size_t)row * DD + col + 32] = acc2[i];
      C[(size_t)row * DD + col + 48] = acc3[i];
    }
  }
}

// ---------------- row softmax over 512 cols (grid = #rows, block 256) ----------------
__global__ void softmax512(const float* __restrict__ in, float* __restrict__ out) {
  __shared__ float redm[8];
  __shared__ float reds[8];
  const int row = blockIdx.x, tid = threadIdx.x;
  const float v0 = in[(size_t)row * DD + tid];
  const float v1 = in[(size_t)row * DD + 256 + tid];

  float m = fmaxf(v0, v1);
  for (int o = 16; o > 0; o >>= 1) m = fmaxf(m, __shfl_xor(m, o));
  if ((tid & 31) == 0) redm[tid >> 5] = m;
  __syncthreads();
  float mm = redm[0];
#pragma unroll
  for (int i = 1; i < 8; ++i) mm = fmaxf(mm, redm[i]);

  const float e0 = __expf(v0 - mm), e1 = __expf(v1 - mm);
  float s = e0 + e1;
  for (int o = 16; o > 0; o >>= 1) s += __shfl_xor(s, o);
  if ((tid & 31) == 0) reds[tid >> 5] = s;
  __syncthreads();
  float ss = 0.f;
#pragma unroll
  for (int i = 0; i < 8; ++i) ss += reds[i];
  const float inv = 1.0f / ss;

  out[(size_t)row * DD + tid]       = e0 * inv;
  out[(size_t)row * DD + 256 + tid] = e1 * inv;
}

// ---- pack w2 into WMMA B-fragment pair layout:
// out[kpair*1024 + 2*n + p] = w2[n*512 + 2*kpair + p]  (p in {0,1})
// => lane reads float2 {B[k][n], B[k+1][n]} as ONE coalesced b64 load.
__global__ void pack_w2_pairs(const float* __restrict__ in, float* __restrict__ out) {
  const int idx = blockIdx.x * 256 + threadIdx.x;   // output-linear -> coalesced stores
  const int kpair = idx >> 10;
  const int rem   = idx & 1023;
  const int n     = rem >> 1;
  const int k     = (kpair << 1) | (rem & 1);
  out[idx] = in[(size_t)n * DD + k];
}

// ---------------- fused pair net ------------------------------------------------------
// OUT[(b*Mdim+m), n] = relu(xp[b,:] + yp[m,:] + b1) @ w2^T + b2,  n in [0,512)
// w2pair is the pair-packed B layout above. grid.x = totalRows/16, block 256 (8 waves).
__global__ void fused_pair(const float* __restrict__ xp, const float* __restrict__ yp,
                           const float* __restrict__ b1, const float* __restrict__ w2pair,
                           const float* __restrict__ b2, float* __restrict__ out,
                           int Mdim) {
  __shared__ float As[16][DD + 4];     // padded stride: 516 floats -> conflict-free frags
  __shared__ int   bArr[16], mArr[16];

  const int tid   = threadIdx.x;
  const int rBase = blockIdx.x * 16;   // totalRows is a multiple of 16 for both calls

  if (tid < 16) {
    const int r = rBase + tid;
    bArr[tid] = r / Mdim;
    mArr[tid] = r % Mdim;
  }
  __syncthreads();

  // Stage A rows: relu(x + y + b1), 16 x 512 floats, 32 per thread, coalesced loads.
  for (int idx = tid; idx < 16 * DD; idx += 256) {
    const int row = idx >> 9, col = idx & (DD - 1);
    const float v = xp[(size_t)bArr[row] * DD + col] +
                    yp[(size_t)mArr[row] * DD + col] + b1[col];
    As[row][col] = v > 0.f ? v : 0.f;
  }
  __syncthreads();

  const int wave = tid >> 5, lane = tid & 31;
  const int hi   = lane >> 4, lr = lane & 15;
  const int nBase = wave * 64;

  // B: float2 stream, kpair-major; per k-step (kk += 4) kpair advances by 2 -> +1024 v2f.
  const v2f* __restrict__ bp = (const v2f*)w2pair + (size_t)hi * DD + nBase + lr;
  const float* ap = &As[lr][2 * hi];   // LDS: ds_load_b64 per step

  v8f acc0 = {0,0,0,0,0,0,0,0}, acc1 = acc0, acc2 = acc0, acc3 = acc0;

#pragma unroll 4
  for (int kk = 0; kk < DD; kk += 4) {
    v2f a; a.x = ap[0]; a.y = ap[1]; ap += 4;

    v2f b0 = bp[0];
    v2f b1f = bp[16];
    v2f b2f = bp[32];
    v2f b3f = bp[48];
    bp += 2 * DD;

    acc0 = wmma4(a, b0, acc0);
    acc1 = wmma4(a, b1f, acc1);
    acc2 = wmma4(a, b2f, acc2);
    acc3 = wmma4(a, b3f, acc3);
  }

  // Streaming epilogue: output is written once and never read again -> non-temporal
  // stores keep w2/non_proj/indi_proj resident in L2 while 512 MB flows to HBM.
#pragma unroll
  for (int i = 0; i < 8; ++i) {
    const size_t row = (size_t)rBase + i + 8 * hi;
    int col = nBase + lr;
    __builtin_nontemporal_store(acc0[i] + b2[col], &out[row * DD + col]); col += 16;
    __builtin_nontemporal_store(acc1[i] + b2[col], &out[row * DD + col]); col += 16;
    __builtin_nontemporal_store(acc2[i] + b2[col], &out[row * DD + col]); col += 16;
    __builtin_nontemporal_store(acc3[i] + b2[col], &out[row * DD + col]);
  }
}

extern "C" void kernel_launch(void* const* d_in, const int* in_sizes, int n_in,
                              void* d_out, int out_size, void* d_ws, size_t ws_size,
                              hipStream_t stream) {
  (void)in_sizes; (void)n_in; (void)out_size; (void)ws_size;

  const float* nonindi  = (const float*)d_in[0];
  const float* indi     = (const float*)d_in[1];
  const float* q_prompt = (const float*)d_in[2];
  const float* q_wq = (const float*)d_in[3];  const float* q_bq = (const float*)d_in[4];
  const float* q_wk = (const float*)d_in[5];  const float* q_bk = (const float*)d_in[6];
  const float* q_wv = (const float*)d_in[7];  const float* q_bv = (const float*)d_in[8];
  const float* q_wo = (const float*)d_in[9];  const float* q_bo = (const float*)d_in[10];
  // d_in[11..19]: qe_* branch is dead code in the reference (result discarded) -> skipped
  const float* net_w1 = (const float*)d_in[20];
  const float* net_b1 = (const float*)d_in[21];
  const float* net_w2 = (const float*)d_in[22];
  const float* net_b2 = (const float*)d_in[23];

  float* ws = (float*)d_ws;
  float* kmat  = ws;                 // 512*512
  float* vmat  = ws + 262144;        // 512*512
  float* qmat  = ws + 524288;        // 16*512 (rows >= 10 never read: A-loads clamped)
  float* sim   = ws + 532480;        // 16*512
  float* attn  = ws + 540672;        // 16*512
  float* out0  = ws + 548864;        // 16*512
  float* outp  = ws + 557056;        // 16*512
  float* proj2 = ws + 565248;        // 16*512
  float* nonp  = ws + 573440;        // 512*512 : nonindi @ w1a^T
  float* indip = ws + 835584;        // 512*512 : indi @ w1b^T
  float* w2pr  = ws + 1097728;       // 512*512 : net_w2 in pair-packed B-frag layout

  float* fact_out = (float*)d_out;                          // (512,10,512)
  float* ctr_out  = (float*)d_out + (size_t)BB * NPRM * DD; // (512,512,512)

  const dim3 blk(256);

  // w2 -> pair-packed B-fragment layout once
  pack_w2_pairs<<<dim3(1024), blk, 0, stream>>>(net_w2, w2pr);

  // attention on indi
  gemm_at512<<<dim3(1),  blk, 0, stream>>>(q_prompt, NPRM, DD, q_wq, DD, DD, q_bq, 1.0f, qmat);
  gemm_at512<<<dim3(32), blk, 0, stream>>>(indi,     BB,   DD, q_wk, DD, DD, q_bk, 1.0f, kmat);
  gemm_at512<<<dim3(32), blk, 0, stream>>>(indi,     BB,   DD, q_wv, DD, DD, q_bv, 1.0f, vmat);
  gemm_at512<<<dim3(1),  blk, 0, stream>>>(qmat, NPRM, DD, kmat, DD, DD, nullptr, FSCALE, sim);
  softmax512<<<dim3(NPRM), blk, 0, stream>>>(sim, attn);
  gemm_ab512<<<dim3(1),  blk, 0, stream>>>(attn, NPRM, DD, vmat, DD, DD, out0);
  gemm_at512<<<dim3(1),  blk, 0, stream>>>(out0, NPRM, DD, q_wo, DD, DD, q_bo, 1.0f, outp);
  // proj2 = out @ w1b^T  (w1b = net_w1[:, 512:], row stride 1024)
  gemm_at512<<<dim3(1),  blk, 0, stream>>>(outp, NPRM, DD, net_w1 + DD, 2 * DD, DD, nullptr, 1.0f, proj2);

  // projections of the pair inputs (w1a = net_w1[:, :512])
  gemm_at512<<<dim3(32), blk, 0, stream>>>(nonindi, BB, DD, net_w1,      2 * DD, DD, nullptr, 1.0f, nonp);
  gemm_at512<<<dim3(32), blk, 0, stream>>>(indi,    BB, DD, net_w1 + DD, 2 * DD, DD, nullptr, 1.0f, indip);

  // fused relu-pair GEMMs (store-bound; h never materialized)
  fused_pair<<<dim3((BB * NPRM) / 16), blk, 0, stream>>>(nonp, proj2, net_b1, w2pr, net_b2, fact_out, NPRM);
  fused_pair<<<dim3((BB * BB)  / 16), blk, 0, stream>>>(nonp, indip, net_b1, w2pr, net_b2, ctr_out, BB);
}